// unbiasedExponentialSmoother_83124797047438
// MI455X (gfx1250) — compile-verified
//
#include <hip/hip_runtime.h>

// EWMA (unbiased) normalization, sequential over T per channel.
// Shapes fixed by the reference: s (16, 2000, 257, 2) f32.
#define N_        16
#define T_        2000
#define FC_       514            // F*C = 257*2, contiguous fastest dims
#define TT_       16             // time steps per tile
#define NBUF_     10             // LDS ring buffers
#define CHUNK_W_  130            // channels per block (overlapping chunks)
#define PAIRS_    (TT_ * 65)     // 1040 float2 pairs per tile
#define ROWSTRIDE_ 132           // floats per LDS row (528 B, 8B-aligned)
#define TILE_FLOATS_ (TT_ * ROWSTRIDE_)  // 2112 floats = 8448 B per buffer
#define TILE_BYTES_  (TILE_FLOATS_ * 4)
#define THREADS_  160            // 5 waves (wave32)
#define OPS_      7              // async issues per wave per tile (uniform!)
#define NUMTILES_ (T_ / TT_)     // 125
#define WAIT_IMM_ ((NBUF_ - 2) * OPS_)   // 56 <= 63 (ASYNCcnt is 6 bits)

__device__ __forceinline__ void async_wait_tiles() {
#if defined(__has_builtin) && __has_builtin(__builtin_amdgcn_s_wait_asynccnt)
  __builtin_amdgcn_s_wait_asynccnt(WAIT_IMM_);
#else
  asm volatile("s_wait_asynccnt %0" :: "i"(WAIT_IMM_) : "memory");
#endif
}

__device__ __forceinline__ void wg_barrier() {
  asm volatile("s_barrier_signal -1\n\ts_barrier_wait -1" ::: "memory");
}

// Issue one tile (TT_ rows x 130 floats) as PAIRS_ float2 async copies to LDS
// using GVS addressing: uniform 64-bit tile base in an SGPR pair, invariant
// per-lane byte offsets in VGPRs. Every wave issues exactly OPS_ instructions
// so per-wave ASYNCcnt accounting is uniform across all 5 waves.
__device__ __forceinline__ void async_issue_tile(
    const float* __restrict__ sbase,          // uniform tile base (SGPR pair)
    unsigned lbase,                           // uniform LDS ring-buffer base
    const unsigned (&lane_lds)[OPS_],         // invariant per-lane LDS offsets
    const unsigned (&lane_goff)[OPS_]) {      // invariant per-lane glb offsets
#pragma unroll
  for (int r = 0; r < OPS_; ++r) {
    unsigned loff = lbase + lane_lds[r];
    asm volatile("global_load_async_to_lds_b64 %0, %1, %2"
                 :: "v"(loff), "v"(lane_goff[r]), "s"(sbase) : "memory");
  }
}

__global__ __launch_bounds__(THREADS_, 1)
void ewma_norm_kernel(const float* __restrict__ in, float* __restrict__ out) {
  __shared__ float smem[NBUF_ * TILE_FLOATS_];   // 84,480 B static LDS

  const int tid   = (int)threadIdx.x;
  const int n     = (int)blockIdx.x >> 2;
  const int chunk = (int)blockIdx.x & 3;
  const int fc0   = chunk * 128;   // chunks: [0,130) [128,258) [256,386) [384,514)

  const float* gbase = in + (size_t)n * T_ * FC_ + fc0;
  // Low 32 bits of a generic LDS pointer == wave-relative LDS byte address.
  const unsigned smem_base = (unsigned)(unsigned long long)(void*)smem;

  // Precompute the loop-invariant per-lane address components once.
  unsigned lane_lds[OPS_], lane_goff[OPS_];
#pragma unroll
  for (int r = 0; r < OPS_; ++r) {
    int p = r * THREADS_ + tid;
    p = (p < PAIRS_) ? p : (PAIRS_ - 1);   // clamp: duplicate tail pair, benign
    int trow = p / 65;                      // time row within tile
    int col2 = p - trow * 65;               // float2 column
    lane_goff[r] = (unsigned)((trow * FC_ + 2 * col2) * 4);
    lane_lds[r]  = (unsigned)(trow * (ROWSTRIDE_ * 4) + col2 * 8);
  }

  // Prime the pipeline: tiles 0..NBUF_-2 into buffers 0..NBUF_-2.
  // (NBUF_-1)*OPS_ = 63 outstanding == ASYNCcnt max; legal (issue stalls,
  // never overflows).
  for (int k = 0; k < NBUF_ - 1; ++k) {
    async_issue_tile(gbase + (size_t)k * TT_ * FC_,
                     smem_base + (unsigned)(k * TILE_BYTES_),
                     lane_lds, lane_goff);
  }

  // Per-channel recurrence state.
  const float beta = 0.99f, cb = 0.01f, eps = 1e-5f;
  float bt = 1.0f, v1 = 0.0f, v2 = 0.0f, S = 0.0f;

  float* op = out + (size_t)n * T_ * FC_ + fc0 + tid;
  const bool active = (tid < CHUNK_W_);

  for (int i = 0; i < NUMTILES_; ++i) {
    // Tile i complete for this wave once <= (NBUF_-2) tiles remain in flight.
    async_wait_tiles();
    // All waves: tile i resident in LDS; all waves done computing tile i-1.
    wg_barrier();

    // Issue tile i+NBUF_-1 into buffer (i+NBUF_-1)%NBUF_ == (i-1)%NBUF_,
    // whose previous contents (tile i-1) every wave just finished. Past the
    // end, re-issue the last tile ("dummy") so ASYNCcnt stays uniform; the
    // NBUF_-1 dummy targets cover every residue except the last tile's live
    // buffer, each overwriting only an already-computed, never-read-again one.
    {
      int tIss = i + NBUF_ - 1;
      int bIss = tIss % NBUF_;
      if (tIss > NUMTILES_ - 1) tIss = NUMTILES_ - 1;
      async_issue_tile(gbase + (size_t)tIss * TT_ * FC_,     // uniform -> SALU
                       smem_base + (unsigned)(bIss * TILE_BYTES_),
                       lane_lds, lane_goff);
    }

    const int bufo = (i % NBUF_) * TILE_FLOATS_;
    if (active) {
#pragma unroll
      for (int r = 0; r < TT_; ++r) {
        float x = smem[bufo + r * ROWSTRIDE_ + tid];
        bt *= beta;                               // beta^t
        v1 = fmaf(beta, v1, cb * x);              // EWMA numerator
        float a   = 1.0f - bt;
        float inv = __builtin_amdgcn_rcpf(a);     // 1/(1-beta^t)
        inv = inv * (2.0f - a * inv);             // Newton refine
        float v2n = v1 * inv;                     // unbiased mean
        float w   = cb * inv;
        float d   = x - v2n;
        float dp  = x - v2;                       // uses previous v2
        S = fmaf(w, fmaf(d, dp, -S), S);          // (1-w)S + w*d*dp
        S = fmaxf(S, 0.0f);                       // clamp
        float y = d * __builtin_amdgcn_rsqf(S + eps);
        v2 = v2n;
        // Write-once output: keep it out of the caches (NT store).
        __builtin_nontemporal_store(y, op + (size_t)(i * TT_ + r) * FC_);
      }
    }
  }
  // S_ENDPGM performs an implicit wait-idle for the trailing dummy issues.
}

extern "C" void kernel_launch(void* const* d_in, const int* in_sizes, int n_in,
                              void* d_out, int out_size, void* d_ws, size_t ws_size,
                              hipStream_t stream) {
  (void)in_sizes; (void)n_in; (void)out_size; (void)d_ws; (void)ws_size;
  const float* s = (const float*)d_in[0];
  float* y = (float*)d_out;
  dim3 grid(N_ * 4);       // 16 batches x 4 channel chunks
  dim3 block(THREADS_);    // 160 threads = 5 wave32
  ewma_norm_kernel<<<grid, block, 0, stream>>>(s, y);
}